// ConvexMultiHeadAttention_88527865905586
// MI455X (gfx1250) — compile-verified
//
#include <hip/hip_runtime.h>
#include <stdint.h>

typedef __attribute__((ext_vector_type(2))) float v2f;
typedef __attribute__((ext_vector_type(8))) float v8f;
typedef __attribute__((ext_vector_type(4))) unsigned int v4u;
typedef __attribute__((ext_vector_type(8))) int v8i;
typedef __attribute__((ext_vector_type(4))) int v4i;

#define D_MODEL 1024
#define NUM_HEADS 16
#define HEAD_DIM 64
#define SEQ 2048

// LDS offset of a __shared__ object: for flat addresses in the LDS aperture,
// the low 32 bits of the address are the LDS offset (ISA 10.2), so truncation yields the DS byte address.
__device__ __forceinline__ uint32_t lds_addr_of(const void* p) {
    return (uint32_t)(uintptr_t)p;
}

// CDNA5 async DMA: memory -> LDS, 16B per lane, tracked by ASYNCcnt.
__device__ __forceinline__ void async_copy_b128(uint32_t lds_dst, const void* gsrc) {
    asm volatile("global_load_async_to_lds_b128 %0, %1, off"
                 :
                 : "v"(lds_dst), "v"((uint64_t)(uintptr_t)gsrc)
                 : "memory");
}

// CDNA5 Tensor Data Mover: one descriptor-driven DMA of a 2D tile
//   64 rows x 64 f32, global row stride D_MODEL elements,
//   LDS padded 64 payload DW + 4 pad DW per row (= float[68] stride).
// D# per ISA 8.3/8.4: group0 = {flags/count, lds_addr, global_addr, type=2},
// group1 = {data_size=4B, pad_interval=64DW(code 5), pad_amount=4DW(code 3), dims}.
// This toolchain's builtin is the 6-arg form:
//   (uint32x4 g0, int32x8 g1, int32x4, int32x4, int32x8, i32 cpol)
__device__ __forceinline__ void tdm_load_kv_tile(uint32_t lds_byte_addr, const float* gsrc) {
    const uint64_t ga = (uint64_t)(uintptr_t)gsrc;
    v4u g0;
    g0.x = 1u;                                              // count=1 (valid), user mode
    g0.y = lds_byte_addr;                                   // lds_addr
    g0.z = (uint32_t)(ga & 0xFFFFFFFFu);                    // global_addr[31:0]
    g0.w = (uint32_t)((ga >> 32) & 0x01FFFFFFu) | (2u << 30); // addr[56:32] | type=2
    v8i g1;
    g1[0] = (2 << 16) | (1 << 20) | (5 << 22) | (3 << 25);  // 4B elems, pad 64DW -> +4DW
    g1[1] = 0;                                              // abar=0, tensor_dim0[15:0]=0
    g1[2] = (int)(((1u << 20) >> 16) & 0xFFFFu);            // tensor_dim0 = 1<<20 (no OOB)
    g1[3] = (int)((((1u << 20) >> 16) & 0xFFFFu) | (64u << 16)); // tensor_dim1 hi | tile_dim0=64
    g1[4] = 64;                                             // tile_dim1=64, tile_dim2=0 (2D)
    g1[5] = 1024;                                           // tensor_dim0_stride = 1024 elems
    g1[6] = 0;                                              // stride hi | dim1_stride lo
    g1[7] = 0;
    const v4i gz4 = {0, 0, 0, 0};                           // groups 2/3 unused (2D tile)
    const v8i gz8 = {0, 0, 0, 0, 0, 0, 0, 0};
    __builtin_amdgcn_tensor_load_to_lds(g0, g1, gz4, gz4, gz8, 0);
}

// ---------------------------------------------------------------------------
// Kernel 1: x_proj = x @ W + b   (M=4096, N=K=1024), f32 WMMA 16x16x4.
// 128 threads / 4 waves; 64x64 tile; each wave owns a 16x64 strip.
// Double-buffered async LDS fill (4 async b128 per wave per 16-deep K slab).
// ---------------------------------------------------------------------------
__global__ __launch_bounds__(128) void proj_gemm_kernel(
    const float* __restrict__ x, const float* __restrict__ W,
    const float* __restrict__ bias, float* __restrict__ xp)
{
    __shared__ float As[2][64][20];  // As[buf][m][k], stride 80B (16B-aligned, conflict-free)
    __shared__ float Bs[2][16][68];  // Bs[buf][k][n], stride 272B

    const int m0 = blockIdx.x * 64;
    const int n0 = blockIdx.y * 64;
    const int t    = threadIdx.x;
    const int wave = t >> 5;
    const int lane = t & 31;
    const int l16  = lane & 15;
    const int lh   = lane >> 4;

    auto issue_tile = [&](int buf, int k0) {
        #pragma unroll
        for (int p = 0; p < 2; ++p) {             // As: 64x16 f32 = 256 b128 lane-tasks
            int u = p * 128 + t;
            int r = u >> 2, q = u & 3;
            async_copy_b128(lds_addr_of(&As[buf][r][q * 4]),
                            x + (size_t)(m0 + r) * D_MODEL + k0 + q * 4);
        }
        #pragma unroll
        for (int p = 0; p < 2; ++p) {             // Bs: 16x64 f32 = 256 b128 lane-tasks
            int u = p * 128 + t;
            int k = u >> 4, c = u & 15;
            async_copy_b128(lds_addr_of(&Bs[buf][k][c * 4]),
                            W + (size_t)(k0 + k) * D_MODEL + n0 + c * 4);
        }
    };

    v8f acc[4] = {};
    issue_tile(0, 0);

    for (int i = 0; i < D_MODEL / 16; ++i) {
        if (i + 1 < D_MODEL / 16) {
            issue_tile((i + 1) & 1, (i + 1) * 16);            // prefetch next slab
            asm volatile("s_wait_asynccnt 0x4" ::: "memory"); // current slab done (in-order)
        } else {
            asm volatile("s_wait_asynccnt 0x0" ::: "memory");
        }
        __syncthreads();

        const int buf  = i & 1;
        const int mrow = wave * 16 + l16;
        #pragma unroll
        for (int kk = 0; kk < 4; ++kk) {
            v2f a;  // A frag (m,k): lane = m + 16*(k>=2), vgpr = k&1
            a.x = As[buf][mrow][kk * 4 + 2 * lh + 0];
            a.y = As[buf][mrow][kk * 4 + 2 * lh + 1];
            #pragma unroll
            for (int n = 0; n < 4; ++n) {
                v2f bf; // B frag (k,n): lane = n + 16*(k>=2), vgpr = k&1
                bf.x = Bs[buf][kk * 4 + 2 * lh + 0][n * 16 + l16];
                bf.y = Bs[buf][kk * 4 + 2 * lh + 1][n * 16 + l16];
                acc[n] = __builtin_amdgcn_wmma_f32_16x16x4_f32(
                    false, a, false, bf, (short)0, acc[n], false, false);
            }
        }
        __syncthreads();   // all waves done with buf before it is refilled
    }

    #pragma unroll
    for (int n = 0; n < 4; ++n) {
        const int ncol = n0 + n * 16 + l16;
        const float bv = bias[ncol];
        #pragma unroll
        for (int v = 0; v < 8; ++v) {   // C/D frag (m,n): lane = n + 16*(m>=8), vgpr = m&7
            const int m = m0 + wave * 16 + v + 8 * lh;
            xp[(size_t)m * D_MODEL + ncol] = acc[n][v] + bv;
        }
    }
}

// ---------------------------------------------------------------------------
// Kernel 2: streaming convex-softmax attention for one (b, h, 64-query tile).
// d_q*d_k folded into Q tile; d_v applied in epilogue -> K and V share ONE raw
// x_proj LDS tile. KV tiles delivered by the Tensor Data Mover (double-
// buffered; wave 0 issues, TENSORcnt + workgroup barrier publish to all).
// z clipped to [-15,15] => no running max needed; rowsum accumulated streaming.
// ---------------------------------------------------------------------------
__global__ __launch_bounds__(128) void attn_kernel(
    const float* __restrict__ xp, const float* __restrict__ dq,
    const float* __restrict__ dk, const float* __restrict__ dv,
    float* __restrict__ out)
{
    __shared__ float Qs[64][68];      // scaled Q tile; reused as per-wave P strips
    __shared__ float KVs[2][64][68];  // raw x_proj rows: serves as both K and V
    __shared__ float red[4][16][16];
    __shared__ float rsum[64];

    const int qtile = blockIdx.x;     // 0..31
    const int bh    = blockIdx.y;     // 0..31
    const int bb    = bh >> 4;
    const int h     = bh & 15;
    const int t     = threadIdx.x;
    const int wave  = t >> 5;
    const int lane  = t & 31;
    const int l16   = lane & 15;
    const int lh    = lane >> 4;
    const int hc0   = h * HEAD_DIM;

    const size_t qbase  = (size_t)bb * SEQ + (size_t)qtile * 64;
    const size_t kvrow0 = (size_t)bb * SEQ;

    // First KV tile DMA overlaps with the Q load below (one TDM op, wave 0 only)
    if (wave == 0) {
        tdm_load_kv_tile(lds_addr_of(&KVs[0][0][0]),
                         xp + (kvrow0 + 0) * D_MODEL + hc0);
    }

    // Q tile scaled by folded d_q*d_k
    #pragma unroll
    for (int p = 0; p < 8; ++p) {
        int r = p * 8 + (t >> 4);
        int c = (t & 15) * 4;
        float4 v  = *(const float4*)(xp + (qbase + r) * D_MODEL + hc0 + c);
        float4 sq = *(const float4*)(dq + hc0 + c);
        float4 sk = *(const float4*)(dk + hc0 + c);
        Qs[r][c + 0] = v.x * (sq.x * sk.x); Qs[r][c + 1] = v.y * (sq.y * sk.y);
        Qs[r][c + 2] = v.z * (sq.z * sk.z); Qs[r][c + 3] = v.w * (sq.w * sk.w);
    }
    __syncthreads();

    const int mrow = wave * 16 + l16;
    v2f qa[16];                        // Q A-fragments live in registers all sweep
    #pragma unroll
    for (int kk = 0; kk < 16; ++kk) {
        qa[kk].x = Qs[mrow][kk * 4 + 2 * lh + 0];
        qa[kk].y = Qs[mrow][kk * 4 + 2 * lh + 1];
    }

    v8f oacc[4] = {};
    v8f rs = {};

    for (int kb = 0; kb < SEQ / 64; ++kb) {
        if (wave == 0) {
            if (kb + 1 < SEQ / 64) {   // prefetch next KV tile, then retire current
                tdm_load_kv_tile(lds_addr_of(&KVs[(kb + 1) & 1][0][0]),
                                 xp + (kvrow0 + (size_t)(kb + 1) * 64) * D_MODEL + hc0);
                __builtin_amdgcn_s_wait_tensorcnt(1);   // TDM in-order: tile kb done
            } else {
                __builtin_amdgcn_s_wait_tensorcnt(0);
            }
        }
        __syncthreads();               // publish wave-0's completed DMA to all waves
        const int buf = kb & 1;

        // S = Q * K^T, elementwise convex numerator, spill P to own LDS strip
        #pragma unroll
        for (int n = 0; n < 4; ++n) {
            v8f s = {};
            #pragma unroll
            for (int kk = 0; kk < 16; ++kk) {
                v2f kf;  // B frag of K^T: (k=dh, n=kv)
                kf.x = KVs[buf][n * 16 + l16][kk * 4 + 2 * lh + 0];
                kf.y = KVs[buf][n * 16 + l16][kk * 4 + 2 * lh + 1];
                s = __builtin_amdgcn_wmma_f32_16x16x4_f32(
                    false, qa[kk], false, kf, (short)0, s, false, false);
            }
            #pragma unroll
            for (int v = 0; v < 8; ++v) {
                float z  = fminf(fmaxf(s[v], -15.0f), 15.0f);
                float nm = __expf(z - 1.0f) + 0.1f * z;        // exp(z-R) + LAM*z
                nm = fmaxf(nm, 0.0f);                          // relu
                rs[v] += nm;                                   // per-lane rowsum partial
                Qs[wave * 16 + v + 8 * lh][n * 16 + l16] = nm; // P (own rows only)
            }
        }

        // oacc += P @ V_raw   (same LDS tile doubles as V)
        #pragma unroll
        for (int kk = 0; kk < 16; ++kk) {
            v2f pa;
            pa.x = Qs[mrow][kk * 4 + 2 * lh + 0];
            pa.y = Qs[mrow][kk * 4 + 2 * lh + 1];
            #pragma unroll
            for (int nd = 0; nd < 4; ++nd) {
                v2f vf;  // B frag of V: (k=kv local, n=dh)
                vf.x = KVs[buf][kk * 4 + 2 * lh + 0][nd * 16 + l16];
                vf.y = KVs[buf][kk * 4 + 2 * lh + 1][nd * 16 + l16];
                oacc[nd] = __builtin_amdgcn_wmma_f32_16x16x4_f32(
                    false, pa, false, vf, (short)0, oacc[nd], false, false);
            }
        }
        __syncthreads();   // everyone done with buf before its refill (issued next iter)
    }

    // Cross-lane rowsum reduction through per-wave LDS region
    #pragma unroll
    for (int v = 0; v < 8; ++v)
        red[wave][v + 8 * lh][l16] = rs[v];
    if (lane < 16) {
        float a = 0.0f;
        #pragma unroll
        for (int j = 0; j < 16; ++j) a += red[wave][lane][j];
        rsum[wave * 16 + lane] = a;
    }

    // Epilogue: apply d_v (deferred V scale), divide by rowsum+eps, store merged
    #pragma unroll
    for (int nd = 0; nd < 4; ++nd) {
        const float dvv = dv[hc0 + nd * 16 + l16];
        #pragma unroll
        for (int v = 0; v < 8; ++v) {
            const int m_local = v + 8 * lh;
            const float denom = rsum[wave * 16 + m_local] + 1e-9f;
            const size_t off = (qbase + (size_t)(wave * 16 + m_local)) * D_MODEL
                             + hc0 + nd * 16 + l16;
            out[off] = (oacc[nd][v] * dvv) / denom;
        }
    }
}

// ---------------------------------------------------------------------------
extern "C" void kernel_launch(void* const* d_in, const int* in_sizes, int n_in,
                              void* d_out, int out_size, void* d_ws, size_t ws_size,
                              hipStream_t stream) {
    const float* x  = (const float*)d_in[0];
    const float* W  = (const float*)d_in[1];
    const float* b  = (const float*)d_in[2];
    const float* dq = (const float*)d_in[3];
    const float* dk = (const float*)d_in[4];
    const float* dv = (const float*)d_in[5];
    float* out = (float*)d_out;
    float* xp  = (float*)d_ws;   // x_proj: 4096*1024 f32 = 16 MiB scratch

    dim3 g1(4096 / 64, D_MODEL / 64), b1(128);
    proj_gemm_kernel<<<g1, b1, 0, stream>>>(x, W, b, xp);

    dim3 g2(SEQ / 64, 2 * NUM_HEADS), b2(128);
    attn_kernel<<<g2, b2, 0, stream>>>(xp, dq, dk, dv, out);

    (void)in_sizes; (void)n_in; (void)out_size; (void)ws_size;
}